// attention_36747740184840
// MI455X (gfx1250) — compile-verified
//
#include <hip/hip_runtime.h>
#include <math.h>

// Problem constants (match reference)
#define BATCH   64
#define TLEN    2048
#define HDIM    512
#define OUT_DIM 128

#define CHUNKS  16
#define ROWS_PER_CHUNK (TLEN / CHUNKS)   // 128
#define K2_THREADS 256
#define K2_WAVES   8                      // wave32 -> 8 waves / 256 threads

typedef __attribute__((ext_vector_type(2))) float v2f;
typedef __attribute__((ext_vector_type(8))) float v8f;

// ---------------------------------------------------------------------------
// Kernel 1: U[b][h] = sum_k W_score[h][k] * hidden[b][T-1][k]
// GEMM M=HDIM(512) x N=BATCH(64) x K=HDIM(512) with V_WMMA_F32_16X16X4_F32.
// One 16x16 D-tile per wave32.  Fragment layouts per CDNA5 ISA 7.12.2:
//   A 16x4 : lanes 0-15 -> {K=k,k+1}, lanes 16-31 -> {K=k+2,k+3}, M = lane&15
//   B 4x16 : v0 = rows {k, k+2}, v1 = rows {k+1, k+3}, N = lane&15
//   C/D    : VGPR v = row (m0 + v + 8*(lane>>4)), col = m0 is h, n0 is b
// Output stored transposed: Ut[b*HDIM + h] so k2 reads it contiguously.
// ---------------------------------------------------------------------------
__global__ __launch_bounds__(32) void k1_u_gemm(const float* __restrict__ Wscore,
                                                const float* __restrict__ hidden,
                                                float* __restrict__ Ut) {
    const int lane = threadIdx.x;
    const int half = lane >> 4;
    const int r    = lane & 15;
    const int m0   = blockIdx.x * 16;   // h tile
    const int n0   = blockIdx.y * 16;   // b tile
    const size_t last_row = (size_t)(TLEN - 1) * HDIM;

    v8f c = {};
    for (int k = 0; k < HDIM; k += 4) {
        const int ka = k + 2 * half;
        v2f a, b;
        a.x = Wscore[(size_t)(m0 + r) * HDIM + ka];
        a.y = Wscore[(size_t)(m0 + r) * HDIM + ka + 1];
        // B[k][n] = h_t[n][k] = hidden[n, T-1, k]
        const size_t hb = (size_t)(n0 + r) * TLEN * HDIM + last_row;
        b.x = hidden[hb + ka];
        b.y = hidden[hb + ka + 1];
        c = __builtin_amdgcn_wmma_f32_16x16x4_f32(false, a, false, b,
                                                  (short)0, c, false, false);
    }
    for (int v = 0; v < 8; ++v) {
        const int m = m0 + v + 8 * half;              // h index
        Ut[(size_t)(n0 + r) * HDIM + m] = c[v];       // [b][h]
    }
}

// ---------------------------------------------------------------------------
// Kernel 2: streaming fused score + online-softmax + weighted context.
// grid = (BATCH, CHUNKS), block = 256. Each wave processes 16 rows of 512
// floats (float4 / b128 loads), keeps the row in registers for both the dot
// product and the weighted accumulation -> hidden is read exactly once.
// Emits per-chunk partial (m, Z, ctx[512]) for the LSE merge in k3.
// ---------------------------------------------------------------------------
__global__ __launch_bounds__(K2_THREADS) void k2_stream(const float* __restrict__ hidden,
                                                        const float* __restrict__ Ut,
                                                        float* __restrict__ cpart,
                                                        float* __restrict__ mpart,
                                                        float* __restrict__ zpart) {
    __shared__ float u_lds[HDIM];
    __shared__ float wctx[K2_WAVES][HDIM];
    __shared__ float wm[K2_WAVES];
    __shared__ float wz[K2_WAVES];

    const int b     = blockIdx.x;
    const int chunk = blockIdx.y;
    const int tid   = threadIdx.x;
    const int lane  = tid & 31;
    const int wave  = tid >> 5;

    for (int h = tid; h < HDIM; h += K2_THREADS)
        u_lds[h] = Ut[(size_t)b * HDIM + h];
    __syncthreads();

    float m = -1e30f, Z = 0.0f;
    float ctx[16];
#pragma unroll
    for (int i = 0; i < 16; ++i) ctx[i] = 0.0f;

    const float* base = hidden + (size_t)b * TLEN * HDIM;
    const int t0 = chunk * ROWS_PER_CHUNK;

    for (int i = 0; i < ROWS_PER_CHUNK / K2_WAVES; ++i) {
        const int t = t0 + wave + K2_WAVES * i;
        const float* row = base + (size_t)t * HDIM;
        // stage a future row toward L2 (global_prefetch_b8)
        __builtin_prefetch(row + (size_t)K2_WAVES * HDIM + 4 * lane, 0, 1);

        float4 r4[4];
        float s = 0.0f;
#pragma unroll
        for (int j = 0; j < 4; ++j) {
            r4[j] = ((const float4*)row)[lane + 32 * j];   // h = 4*lane + 128*j ..+3
            const int h = 4 * lane + 128 * j;
            s += r4[j].x * u_lds[h]     + r4[j].y * u_lds[h + 1]
               + r4[j].z * u_lds[h + 2] + r4[j].w * u_lds[h + 3];
        }
        // wave32 butterfly reduction -> every lane holds the full score
#pragma unroll
        for (int off = 16; off > 0; off >>= 1)
            s += __shfl_xor(s, off, 32);

        const float m_new = fmaxf(m, s);
        const float scale = __expf(m - m_new);
        const float w     = __expf(s - m_new);
        Z = Z * scale + w;
#pragma unroll
        for (int j = 0; j < 4; ++j) {
            ctx[4 * j + 0] = ctx[4 * j + 0] * scale + w * r4[j].x;
            ctx[4 * j + 1] = ctx[4 * j + 1] * scale + w * r4[j].y;
            ctx[4 * j + 2] = ctx[4 * j + 2] * scale + w * r4[j].z;
            ctx[4 * j + 3] = ctx[4 * j + 3] * scale + w * r4[j].w;
        }
        m = m_new;
    }

#pragma unroll
    for (int j = 0; j < 4; ++j) {
        const int h = 4 * lane + 128 * j;
        wctx[wave][h + 0] = ctx[4 * j + 0];
        wctx[wave][h + 1] = ctx[4 * j + 1];
        wctx[wave][h + 2] = ctx[4 * j + 2];
        wctx[wave][h + 3] = ctx[4 * j + 3];
    }
    if (lane == 0) { wm[wave] = m; wz[wave] = Z; }
    __syncthreads();

    // merge the 8 wave partials (redundantly per thread; trivial cost)
    float mb = -1e30f;
    for (int w2 = 0; w2 < K2_WAVES; ++w2) mb = fmaxf(mb, wm[w2]);
    float zb = 0.0f;
    for (int w2 = 0; w2 < K2_WAVES; ++w2) zb += wz[w2] * __expf(wm[w2] - mb);

    float* cp = cpart + ((size_t)b * CHUNKS + chunk) * HDIM;
    for (int h = tid; h < HDIM; h += K2_THREADS) {
        float acc = 0.0f;
        for (int w2 = 0; w2 < K2_WAVES; ++w2)
            acc += wctx[w2][h] * __expf(wm[w2] - mb);
        cp[h] = acc;
    }
    if (tid == 0) {
        mpart[b * CHUNKS + chunk] = mb;
        zpart[b * CHUNKS + chunk] = zb;
    }
}

// ---------------------------------------------------------------------------
// Kernel 3: per-batch LSE merge of chunk partials; builds A2 = [context|h_t].
// grid = BATCH, block = 512 (one thread per h).
// ---------------------------------------------------------------------------
__global__ __launch_bounds__(512) void k3_combine(const float* __restrict__ hidden,
                                                  const float* __restrict__ cpart,
                                                  const float* __restrict__ mpart,
                                                  const float* __restrict__ zpart,
                                                  float* __restrict__ A2) {
    __shared__ float cm[CHUNKS], cz[CHUNKS];
    const int b = blockIdx.x, tid = threadIdx.x;
    if (tid < CHUNKS) {
        cm[tid] = mpart[b * CHUNKS + tid];
        cz[tid] = zpart[b * CHUNKS + tid];
    }
    __syncthreads();
    float mg = -1e30f;
    for (int c = 0; c < CHUNKS; ++c) mg = fmaxf(mg, cm[c]);
    float zg = 0.0f;
    for (int c = 0; c < CHUNKS; ++c) zg += cz[c] * __expf(cm[c] - mg);
    const float inv = 1.0f / zg;

    const int h = tid;
    float acc = 0.0f;
    for (int c = 0; c < CHUNKS; ++c)
        acc += cpart[((size_t)b * CHUNKS + c) * HDIM + h] * __expf(cm[c] - mg);
    A2[(size_t)b * (2 * HDIM) + h]        = acc * inv;                       // context
    A2[(size_t)b * (2 * HDIM) + HDIM + h] =
        hidden[((size_t)b * TLEN + (TLEN - 1)) * HDIM + h];                  // h_t
}

// ---------------------------------------------------------------------------
// Kernel 4: out = tanh(A2(64x1024) @ W_out(1024x128)) via WMMA f32 16x16x4.
// grid = (4, 8) tiles, one wave per tile.
// ---------------------------------------------------------------------------
__global__ __launch_bounds__(32) void k4_out_gemm(const float* __restrict__ A2,
                                                  const float* __restrict__ Wout,
                                                  float* __restrict__ out) {
    const int lane = threadIdx.x;
    const int half = lane >> 4;
    const int r    = lane & 15;
    const int m0   = blockIdx.x * 16;   // batch tile
    const int n0   = blockIdx.y * 16;   // out-feature tile

    v8f c = {};
    for (int k = 0; k < 2 * HDIM; k += 4) {
        const int ka = k + 2 * half;
        v2f a, b;
        a.x = A2[(size_t)(m0 + r) * (2 * HDIM) + ka];
        a.y = A2[(size_t)(m0 + r) * (2 * HDIM) + ka + 1];
        b.x = Wout[(size_t)ka * OUT_DIM + n0 + r];
        b.y = Wout[(size_t)(ka + 1) * OUT_DIM + n0 + r];
        c = __builtin_amdgcn_wmma_f32_16x16x4_f32(false, a, false, b,
                                                  (short)0, c, false, false);
    }
    for (int v = 0; v < 8; ++v) {
        const int m = m0 + v + 8 * half;
        out[(size_t)m * OUT_DIM + n0 + r] = tanhf(c[v]);
    }
}

// ---------------------------------------------------------------------------
extern "C" void kernel_launch(void* const* d_in, const int* in_sizes, int n_in,
                              void* d_out, int out_size, void* d_ws, size_t ws_size,
                              hipStream_t stream) {
    const float* hidden = (const float*)d_in[0];   // (B,T,H)
    const float* Wscore = (const float*)d_in[1];   // (H,H)
    const float* Wout   = (const float*)d_in[2];   // (2H,OUT)
    float* out = (float*)d_out;                    // (B,OUT)

    // workspace layout (floats): ~2.44 MB total
    float* ws    = (float*)d_ws;
    float* Ut    = ws;                                   // BATCH*HDIM
    float* cpart = Ut    + (size_t)BATCH * HDIM;         // BATCH*CHUNKS*HDIM
    float* mpart = cpart + (size_t)BATCH * CHUNKS * HDIM;// BATCH*CHUNKS
    float* zpart = mpart + (size_t)BATCH * CHUNKS;       // BATCH*CHUNKS
    float* A2    = zpart + (size_t)BATCH * CHUNKS;       // BATCH*2*HDIM

    k1_u_gemm<<<dim3(HDIM / 16, BATCH / 16), 32, 0, stream>>>(Wscore, hidden, Ut);
    k2_stream<<<dim3(BATCH, CHUNKS), K2_THREADS, 0, stream>>>(hidden, Ut,
                                                              cpart, mpart, zpart);
    k3_combine<<<dim3(BATCH), 512, 0, stream>>>(hidden, cpart, mpart, zpart, A2);
    k4_out_gemm<<<dim3(BATCH / 16, OUT_DIM / 16), 32, 0, stream>>>(A2, Wout, out);
}